// FourierLayer_13993003450474
// MI455X (gfx1250) — compile-verified
//
#include <hip/hip_runtime.h>
#include <math.h>

// ---------------------------------------------------------------------------
// FourierLayer for MI455X (gfx1250): DFT + synthesis as f32 WMMA GEMMs.
// Shapes: b=8, t=1024, d=512 -> channels M=4096, K=1024, freqs 1..511,
// T = 1024+256 = 1280 output samples per channel.
// ---------------------------------------------------------------------------

typedef float v2f __attribute__((ext_vector_type(2)));
typedef float v8f __attribute__((ext_vector_type(8)));

#define NB 8          // batch
#define ND 512        // channels per batch
#define NT 1024       // time samples
#define NC 4096       // NB*ND  (GEMM M)
#define NFREQ 511     // kept rfft bins 1..511
#define NPAD 1024     // padded K (2*511 rounded up to 1024)
#define TOUT 1280     // t + PRED_LEN
#define TOPK 32

// Device-pass-only WMMA wrapper: the host pass parses __global__ bodies too,
// and __builtin_amdgcn_* does not exist for x86 — hide it like rocWMMA does.
static __device__ __forceinline__ v8f wmma_f32_16x16x4(v2f a, v2f b, v8f c) {
#if defined(__HIP_DEVICE_COMPILE__)
#if __has_builtin(__builtin_amdgcn_wmma_f32_16x16x4_f32)
  return __builtin_amdgcn_wmma_f32_16x16x4_f32(false, a, false, b, (short)0, c,
                                               false, false);
#else
#error "gfx1250 device pass: __builtin_amdgcn_wmma_f32_16x16x4_f32 not available"
#endif
#else
  return c; // host placeholder, never executed
#endif
}

static __device__ __forceinline__ float twiddle(int prod, int want_sin) {
  // cos/sin(2*pi*prod/1024) with exact integer phase reduction (prod < 2^24)
  int m = prod & 1023;
  float th = (float)m * 6.135923151542565e-3f; // 2*pi/1024
  return want_sin ? sinf(th) : cosf(th);
}

// ---------------------------------------------------------------------------
// Basis generation
// B1: (t, q) ld=1024.  q<511: cos(2pi(q+1)t/1024); q<1022: sin(2pi(q-510)t/1024); else 0
__global__ void gen_b1(float* __restrict__ B1) {
  int idx = blockIdx.x * 256 + threadIdx.x;      // < 1024*1024
  int t = idx >> 10, q = idx & 1023;
  float v = 0.0f;
  if (q < NFREQ)            v = twiddle(t * (q + 1), 0);
  else if (q < 2 * NFREQ)   v = twiddle(t * (q - 510), 1);
  B1[idx] = v;
}

// B2: (q, t') ld=1280.  q<511: cos(2pi(q+1)t'/1024); q<1022: sin; else 0
__global__ void gen_b2(float* __restrict__ B2) {
  int idx = blockIdx.x * 256 + threadIdx.x;      // < 1024*1280
  int q = idx / TOUT, tt = idx - q * TOUT;
  float v = 0.0f;
  if (q < NFREQ)            v = twiddle(tt * (q + 1), 0);
  else if (q < 2 * NFREQ)   v = twiddle(tt * (q - 510), 1);
  B2[idx] = v;
}

// ---------------------------------------------------------------------------
// Standardize: per channel c=(b,d) compute mean/std(ddof=1), write clipped
// signal K-major: xsT[t*4096 + c]  (coalesced reads & writes).
__global__ void standardize(const float* __restrict__ x, float* __restrict__ xsT,
                            float* __restrict__ meanv, float* __restrict__ stdv) {
  int d = blockIdx.x * 256 + threadIdx.x;        // 0..511
  int b = blockIdx.y;
  const float* xp = x + (size_t)b * NT * ND + d;
  float sum = 0.0f;
#pragma unroll 4
  for (int t = 0; t < NT; ++t) sum += xp[(size_t)t * ND];
  float mean = sum * (1.0f / (float)NT);
  float var = 0.0f;
#pragma unroll 4
  for (int t = 0; t < NT; ++t) { float dv = xp[(size_t)t * ND] - mean; var += dv * dv; }
  float sd = sqrtf(var * (1.0f / (float)(NT - 1))) + 1e-8f;
  int c = b * ND + d;
  meanv[c] = mean;
  stdv[c] = sd;
#pragma unroll 4
  for (int t = 0; t < NT; ++t) {
    float v = (xp[(size_t)t * ND] - mean) / sd;
    v = fminf(fmaxf(v, -2.0f), 2.0f) + 1e-9f;
    xsT[(size_t)t * NC + c] = v;
  }
}

// ---------------------------------------------------------------------------
// f32 WMMA GEMM, A supplied K-major (AT: K x 4096), B: K x N, C: 4096 x N.
// Wave computes a 16(M) x 64(N) macro-tile via V_WMMA_F32_16X16X4_F32.
// Block = 128 threads = 4 waves stacked along M.
__global__ __launch_bounds__(128) void wmma_gemm_kt(const float* __restrict__ AT,
                                                    const float* __restrict__ Bm,
                                                    float* __restrict__ Cm,
                                                    int N, int K) {
  int lane = threadIdx.x & 31;
  int wave = threadIdx.x >> 5;
  int m0 = (blockIdx.x * 4 + wave) << 4;
  int n0 = blockIdx.y << 6;
  int l15 = lane & 15;
  int kphase = (lane >> 4) << 1;                 // 0 or 2 (A/B fragment K phase)

  v8f acc[4] = {};

  const float* ap = AT + (size_t)kphase * NC + (m0 + l15);
  const float* bp = Bm + (size_t)kphase * N + (n0 + l15);

  for (int k = 0; k < K; k += 4) {
    v2f a;  a.x  = ap[0];  a.y  = ap[NC];
    v2f b0; b0.x = bp[0];  b0.y = bp[N];
    v2f b1; b1.x = bp[16]; b1.y = bp[N + 16];
    v2f b2; b2.x = bp[32]; b2.y = bp[N + 32];
    v2f b3; b3.x = bp[48]; b3.y = bp[N + 48];
    acc[0] = wmma_f32_16x16x4(a, b0, acc[0]);
    acc[1] = wmma_f32_16x16x4(a, b1, acc[1]);
    acc[2] = wmma_f32_16x16x4(a, b2, acc[2]);
    acc[3] = wmma_f32_16x16x4(a, b3, acc[3]);
    ap += (size_t)4 * NC;
    bp += (size_t)4 * N;
  }

  int rbase = m0 + ((lane >> 4) << 3);           // C layout: VGPR r -> row r / r+8
#pragma unroll
  for (int s = 0; s < 4; ++s)
#pragma unroll
    for (int r = 0; r < 8; ++r)
      Cm[(size_t)(rbase + r) * N + (n0 + s * 16 + l15)] = acc[s][r];
}

// ---------------------------------------------------------------------------
// Per-channel top-32 selection + masked synthesis coefficients.
// One wave per channel. Fcat[c, j] = Re_j (j<511), Fcat[c, 511+j] = sum x*sin
// (= -Im_j).  Writes C2T K-major: row j = cos coeff, row 511+j = sin coeff.
__global__ __launch_bounds__(32) void topk_coeff(const float* __restrict__ Fcat,
                                                 float* __restrict__ C2T) {
  int c = blockIdx.x;
  int lane = threadIdx.x;
  __shared__ float ampS[512];
  const float* Fr = Fcat + (size_t)c * NPAD;

  float re[16], si[16];
#pragma unroll
  for (int i = 0; i < 16; ++i) {
    int j = lane + 32 * i;
    float r = 0.0f, s = 0.0f, a = -1.0f;
    if (j < NFREQ) { r = Fr[j]; s = Fr[NFREQ + j]; a = sqrtf(r * r + s * s); }
    re[i] = r; si[i] = s;
    ampS[i * 32 + lane] = a;
  }
  __syncthreads();

  unsigned sel = 0;
  for (int it = 0; it < TOPK; ++it) {
    float bv = -4.0f; int bj = 1 << 30;
#pragma unroll
    for (int i = 0; i < 16; ++i) {
      float v = ampS[i * 32 + lane];
      int j = lane + 32 * i;
      if (v > bv) { bv = v; bj = j; }            // j increases -> keeps lowest index on ties
    }
    for (int off = 16; off; off >>= 1) {
      float ov = __shfl_xor(bv, off, 32);
      int   oj = __shfl_xor(bj, off, 32);
      if (ov > bv || (ov == bv && oj < bj)) { bv = ov; bj = oj; }
    }
    if (lane == (bj & 31)) {                     // owner marks slot consumed
      ampS[(bj >> 5) * 32 + lane] = -3.0f;
      sel |= (1u << (bj >> 5));
    }
    __syncthreads();
  }

  const float s2 = 2.0f / (1024.0f + 1e-8f);     // 2*|X|/t scaling, split into re/im
#pragma unroll
  for (int i = 0; i < 16; ++i) {
    int j = lane + 32 * i;
    if (j < NFREQ) {
      float on = ((sel >> i) & 1u) ? s2 : 0.0f;
      C2T[(size_t)j * NC + c]           = on * re[i];
      C2T[(size_t)(NFREQ + j) * NC + c] = on * si[i];
    }
  }
  if (lane == 0) {                               // zero the K-padding rows
    C2T[(size_t)1022 * NC + c] = 0.0f;
    C2T[(size_t)1023 * NC + c] = 0.0f;
  }
}

// ---------------------------------------------------------------------------
// Transpose (c, t') -> (b, t', d) with per-channel un-standardization.
__global__ void finalize_out(const float* __restrict__ Y, const float* __restrict__ meanv,
                             const float* __restrict__ stdv, float* __restrict__ out) {
  __shared__ float tile[32][33];
  int b = blockIdx.z;
  int d0 = blockIdx.x * 32, t0 = blockIdx.y * 32;
  int tx = threadIdx.x, ty = threadIdx.y;
  for (int r = ty; r < 32; r += 8)
    tile[r][tx] = Y[(size_t)(b * ND + d0 + r) * TOUT + (t0 + tx)];
  __syncthreads();
  int d = d0 + tx, c = b * ND + d;
  float sd = stdv[c], mn = meanv[c];
  for (int r = ty; r < 32; r += 8)
    out[((size_t)b * TOUT + (t0 + r)) * ND + d] = tile[tx][r] * sd + mn;
}

// ---------------------------------------------------------------------------
extern "C" void kernel_launch(void* const* d_in, const int* in_sizes, int n_in,
                              void* d_out, int out_size, void* d_ws, size_t ws_size,
                              hipStream_t stream) {
  (void)in_sizes; (void)n_in; (void)out_size; (void)ws_size;
  const float* x = (const float*)d_in[0];
  float* out = (float*)d_out;

  float* w = (float*)d_ws;
  float* xsT   = w;                               //  4,194,304 f (K-major signal)
  float* Fcat  = xsT  + (size_t)NT * NC;          //  4,194,304 f (Re | sum-sin)
  float* C2T   = Fcat + (size_t)NC * NPAD;        //  4,194,304 f (masked coeffs, K-major)
  float* B1    = C2T  + (size_t)NPAD * NC;        //  1,048,576 f (DFT basis)
  float* B2    = B1   + (size_t)NT * NPAD;        //  1,310,720 f (synthesis basis)
  float* meanv = B2   + (size_t)NPAD * TOUT;      //  4096 f
  float* stdv  = meanv + NC;                      //  4096 f
  float* Y     = w;                               //  5,242,880 f, aliases dead xsT/Fcat
  // total workspace: ~59.8 MB

  gen_b1<<<(NT * NPAD) / 256, 256, 0, stream>>>(B1);
  gen_b2<<<(NPAD * TOUT) / 256, 256, 0, stream>>>(B2);
  standardize<<<dim3(ND / 256, NB), 256, 0, stream>>>(x, xsT, meanv, stdv);
  // DFT: Fcat(4096 x 1024) = xs(4096 x 1024) @ B1(1024 x 1024)
  wmma_gemm_kt<<<dim3(NC / 64, NPAD / 64), 128, 0, stream>>>(xsT, B1, Fcat, NPAD, NT);
  topk_coeff<<<NC, 32, 0, stream>>>(Fcat, C2T);
  // Synthesis: Y(4096 x 1280) = C2(4096 x 1024) @ B2(1024 x 1280)
  wmma_gemm_kt<<<dim3(NC / 64, TOUT / 64), 128, 0, stream>>>(C2T, B2, Y, TOUT, NPAD);
  finalize_out<<<dim3(ND / 32, TOUT / 32, NB), dim3(32, 8), 0, stream>>>(Y, meanv, stdv, out);
}